// GNN_88613765251253
// MI455X (gfx1250) — compile-verified
//
#include <hip/hip_runtime.h>

typedef float v2f __attribute__((ext_vector_type(2)));
typedef float v8f __attribute__((ext_vector_type(8)));

// ---------------------------------------------------------------------------
// Small utility kernels
// ---------------------------------------------------------------------------
__global__ void k_fill(float* __restrict__ p, float v, int n) {
    int i = blockIdx.x * blockDim.x + threadIdx.x;
    if (i < n) p[i] = v;
}

// deg[dst] += 1 per edge (deg pre-initialized to 1.0 == self loop)
__global__ void k_degree(const long long* __restrict__ dst,
                         float* __restrict__ deg, int E) {
    int e = blockIdx.x * blockDim.x + threadIdx.x;
    if (e < E) atomicAdd(&deg[(int)dst[e]], 1.0f);
}

// dinv = rsqrt(deg)   (deg >= 1 always, matches reference's guarded rsqrt)
__global__ void k_rsqrt(float* __restrict__ d, int n) {
    int i = blockIdx.x * blockDim.x + threadIdx.x;
    if (i < n) d[i] = rsqrtf(d[i]);
}

// per-edge symmetric norm = dinv[src] * dinv[dst]
__global__ void k_norm(const long long* __restrict__ src,
                       const long long* __restrict__ dst,
                       const float* __restrict__ dinv,
                       float* __restrict__ enorm, int E) {
    int e = blockIdx.x * blockDim.x + threadIdx.x;
    if (e < E) enorm[e] = dinv[(int)src[e]] * dinv[(int)dst[e]];
}

// ---------------------------------------------------------------------------
// Strip-GEMM:  H[N,C] = A[N,K] @ W[K,C]
// One wave32 computes a full 16xC strip: the A fragment is loaded once per
// k-step and reused across C/16 accumulators (V_WMMA_F32_16X16X4_F32).
// Adjacent WMMAs hit different accumulators -> no RAW hazard NOPs.
// N % 16 == 0, K % 4 == 0, C % 16 == 0.
// ---------------------------------------------------------------------------
template <int K, int C>
__global__ void k_gemm_strip(const float* __restrict__ A,
                             const float* __restrict__ W,
                             float* __restrict__ H, int tilesM) {
    constexpr int TN = C / 16;
    const int wid = blockIdx.x * (blockDim.x >> 5) + (threadIdx.x >> 5);
    if (wid >= tilesM) return;                  // whole-wave: EXEC all-1 inside
    const int lane = threadIdx.x & 31;
    const int half = lane >> 4;                 // 0: lanes 0-15, 1: lanes 16-31
    const int l15  = lane & 15;

    const float* arow = A + (size_t)(wid * 16 + l15) * K;

    v8f acc[TN];
#pragma unroll
    for (int t = 0; t < TN; ++t) acc[t] = (v8f){};

    // A 16x4 f32 layout: VGPR0 = K{0|2}, VGPR1 = K{1|3} (low/high half-wave)
    // B  4x16 f32 layout: mirrored (rows striped across lanes per VGPR)
#pragma unroll
    for (int k0 = 0; k0 < K; k0 += 4) {
        const int ka = k0 + half * 2;
        v2f a;
        a.x = arow[ka];
        a.y = arow[ka + 1];
#pragma unroll
        for (int t = 0; t < TN; ++t) {
            const int col = t * 16 + l15;
            v2f b;
            b.x = W[ka * C + col];
            b.y = W[(ka + 1) * C + col];
            acc[t] = __builtin_amdgcn_wmma_f32_16x16x4_f32(
                /*neg_a=*/false, a, /*neg_b=*/false, b,
                /*c_mod=*/(short)0, acc[t],
                /*reuse_a=*/false, /*reuse_b=*/false);
        }
    }

    // C/D layout: VGPR r holds (M = 8*half + r, N = col)
    float* hbase = H + (size_t)(wid * 16 + half * 8) * C + l15;
#pragma unroll
    for (int t = 0; t < TN; ++t) {
#pragma unroll
        for (int r = 0; r < 8; ++r)
            hbase[(size_t)r * C + t * 16] = acc[t][r];
    }
}

// Last layer: C == 1, K == 16 -> plain per-node dot product
__global__ void k_gemm_col1(const float* __restrict__ A,
                            const float* __restrict__ W,
                            float* __restrict__ H, int K, int N) {
    int i = blockIdx.x * blockDim.x + threadIdx.x;
    if (i >= N) return;
    const float* a = A + (size_t)i * K;
    float s = 0.f;
#pragma unroll 16
    for (int k = 0; k < K; ++k) s += a[k] * W[k];
    H[i] = s;
}

// ---------------------------------------------------------------------------
// Edge scatter: agg[dst, c] += h[src, c] * norm[e]
// Vectorized: one thread = one edge x 4 channels.  One b128 gather of h
// (L2-resident) + 4 no-return global_atomic_add_f32.
// s4 = log2(C/4); total4 = E * C / 4.
// ---------------------------------------------------------------------------
__global__ void k_scatter4(const long long* __restrict__ src,
                           const long long* __restrict__ dst,
                           const float* __restrict__ enorm,
                           const float* __restrict__ h,
                           float* __restrict__ agg,
                           int s4, int total4) {
    int idx = blockIdx.x * blockDim.x + threadIdx.x;
    if (idx >= total4) return;
    const int e  = idx >> s4;
    const int q  = idx & ((1 << s4) - 1);       // which group of 4 channels
    const int cshift = s4 + 2;                  // log2(C)
    const int s = (int)src[e];
    const int d = (int)dst[e];
    const float w = enorm[e];
    const float4 v =
        *reinterpret_cast<const float4*>(h + ((size_t)s << cshift) + q * 4);
    float* p = agg + ((size_t)d << cshift) + q * 4;
    atomicAdd(p + 0, v.x * w);
    atomicAdd(p + 1, v.y * w);
    atomicAdd(p + 2, v.z * w);
    atomicAdd(p + 3, v.w * w);
}

// scalar scatter for the C==1 layer
__global__ void k_scatter1(const long long* __restrict__ src,
                           const long long* __restrict__ dst,
                           const float* __restrict__ enorm,
                           const float* __restrict__ h,
                           float* __restrict__ agg, int E) {
    int e = blockIdx.x * blockDim.x + threadIdx.x;
    if (e >= E) return;
    atomicAdd(&agg[(int)dst[e]], h[(int)src[e]] * enorm[e]);
}

// out = [relu]( agg + h*dinv^2 (self loop) + bias )
__global__ void k_finalize(const float* __restrict__ agg,
                           const float* __restrict__ h,
                           const float* __restrict__ dinv,
                           const float* __restrict__ bias,
                           float* __restrict__ out,
                           int cshift, int total, int relu) {
    int idx = blockIdx.x * blockDim.x + threadIdx.x;
    if (idx >= total) return;
    const int i = idx >> cshift;
    const int c = idx & ((1 << cshift) - 1);
    const float di = dinv[i];
    float v = agg[idx] + h[idx] * di * di + bias[c];
    out[idx] = relu ? fmaxf(v, 0.0f) : v;
}

// ---------------------------------------------------------------------------
// Host-side orchestration (graph-capture safe: everything on `stream`)
// ---------------------------------------------------------------------------
extern "C" void kernel_launch(void* const* d_in, const int* in_sizes, int n_in,
                              void* d_out, int out_size, void* d_ws, size_t ws_size,
                              hipStream_t stream) {
    (void)n_in; (void)out_size; (void)ws_size;

    const float*     x  = (const float*)d_in[0];
    const long long* ei = (const long long*)d_in[1];   // int64 edge_index [2, E]
    const int N = in_sizes[0] / 64;
    const int E = in_sizes[1] / 2;
    const long long* esrc = ei;
    const long long* edst = ei + E;

    const float* Ws[5] = { (const float*)d_in[2], (const float*)d_in[4],
                           (const float*)d_in[6], (const float*)d_in[8],
                           (const float*)d_in[10] };
    const float* Bs[5] = { (const float*)d_in[3], (const float*)d_in[5],
                           (const float*)d_in[7], (const float*)d_in[9],
                           (const float*)d_in[11] };
    const int dims[6] = { 64, 128, 64, 32, 16, 1 };

    float* ws    = (float*)d_ws;
    float* dinv  = ws;                           // [N]
    float* hbuf  = dinv + N;                     // [N*128] (max C)
    float* agg   = hbuf + (size_t)N * 128;       // [N*128]
    float* feat  = agg  + (size_t)N * 128;       // [N*128]
    float* enorm = feat + (size_t)N * 128;       // [E]

    const int T = 256;                           // 8 waves per block (wave32)

    // --- degrees (self loop folded into init=1), dinv, edge norms ---
    k_fill  <<<(N + T - 1) / T, T, 0, stream>>>(dinv, 1.0f, N);
    k_degree<<<(E + T - 1) / T, T, 0, stream>>>(edst, dinv, E);
    k_rsqrt <<<(N + T - 1) / T, T, 0, stream>>>(dinv, N);
    k_norm  <<<(E + T - 1) / T, T, 0, stream>>>(esrc, edst, dinv, enorm, E);

    const int tilesM = N / 16;
    const int wpb    = T / 32;
    const int gemmBlocks = (tilesM + wpb - 1) / wpb;

    const float* in = x;
    for (int l = 0; l < 5; ++l) {
        const int  C    = dims[l + 1];
        const bool last = (l == 4);
        float* out = last ? (float*)d_out : feat;

        // 1) GEMM: hbuf = in @ W  (WMMA strip kernels, templated per layer)
        switch (l) {
            case 0: k_gemm_strip< 64, 128><<<gemmBlocks, T, 0, stream>>>(in, Ws[0], hbuf, tilesM); break;
            case 1: k_gemm_strip<128,  64><<<gemmBlocks, T, 0, stream>>>(in, Ws[1], hbuf, tilesM); break;
            case 2: k_gemm_strip< 64,  32><<<gemmBlocks, T, 0, stream>>>(in, Ws[2], hbuf, tilesM); break;
            case 3: k_gemm_strip< 32,  16><<<gemmBlocks, T, 0, stream>>>(in, Ws[3], hbuf, tilesM); break;
            case 4: k_gemm_col1<<<(N + T - 1) / T, T, 0, stream>>>(in, Ws[4], hbuf, 16, N); break;
        }

        int cshift = 0;
        while ((1 << cshift) < C) ++cshift;

        // 2) zero accumulator
        const int totNC = N * C;
        k_fill<<<(totNC + T - 1) / T, T, 0, stream>>>(agg, 0.0f, totNC);

        // 3) edge scatter-add (L2-resident gathers + f32 atomics)
        if (C >= 4) {
            const int s4     = cshift - 2;
            const int total4 = E * (C >> 2);     // max 51.2M, fits int32
            k_scatter4<<<(total4 + T - 1) / T, T, 0, stream>>>(
                esrc, edst, enorm, hbuf, agg, s4, total4);
        } else {
            k_scatter1<<<(E + T - 1) / T, T, 0, stream>>>(
                esrc, edst, enorm, hbuf, agg, E);
        }

        // 4) self loop + bias + (relu)
        k_finalize<<<(totNC + T - 1) / T, T, 0, stream>>>(
            agg, hbuf, dinv, Bs[l], out, cshift, totNC, last ? 0 : 1);

        in = feat;
    }
}